// MSDeformableAttention_25409026523845
// MI455X (gfx1250) — compile-verified
//
#include <hip/hip_runtime.h>
#include <hip/hip_bf16.h>

// ---------------- problem constants (fixed by the reference) ----------------
#define EMBED    256
#define HEADS    8
#define HEAD_DIM 32
#define LEVELS   4
#define NPTS     4
#define SUMP     16           // LEVELS*NPTS
#define BS       16
#define LQ       300
#define LEN_V    13294        // 100*100 + 50*50 + 25*25 + 13*13
#define M_VAL    (BS * LEN_V) // 212704 rows for the value projection GEMM
#define M_Q      (BS * LQ)    // 4800 rows for query-side GEMMs

typedef float v2f __attribute__((ext_vector_type(2)));
typedef float v8f __attribute__((ext_vector_type(8)));

// ---------------------------------------------------------------------------
// fp32 WMMA GEMM:  C[M,N] = A[M,256] * B[256,N] + bias[N]
// Block = 256 threads = 8 waves.  Each wave computes a 16(M) x 32(N) tile
// with two f32 accumulators; the block covers 128 rows x 32 cols.
// B tile (256x32, 32KB) is staged in LDS, repacked as [k/2][col][2] so each
// B fragment read is one contiguous ds_load_b64.
// A-row indices are CLAMPED (not predicated) so global loads stay branch-free
// and the scheduler can pipeline them across the unroll; only stores guard M.
// V_WMMA_F32_16X16X4_F32 wave32 layouts (per ISA):
//   A 16x4 : lanes 0-15 hold {k,k+1} of row M=lane, lanes 16-31 hold {k+2,k+3}
//   B 4x16 : lanes 0-15 hold rows k,k+1 at col=lane, lanes 16-31 rows k+2,k+3
//   C 16x16: vgpr i -> M=i (lanes 0-15) / M=i+8 (lanes 16-31), N = lane&15
// ---------------------------------------------------------------------------
__global__ __launch_bounds__(256)
void gemm_wmma_f32(const float* __restrict__ A, const float* __restrict__ B,
                   const float* __restrict__ bias, float* __restrict__ C,
                   int M, int N) {
  __shared__ float Blds[256 * 32];   // [kpair 0..127][col 0..31][j 0..1]

  const int tid = threadIdx.x;
  const int n0  = blockIdx.y * 32;

  // Stage B tile: 256 (K) x 32 (N) -> LDS (repacked), coalesced global reads.
  for (int idx = tid; idx < 256 * 32; idx += 256) {
    const int k = idx >> 5;
    const int c = idx & 31;
    Blds[(k >> 1) * 64 + c * 2 + (k & 1)] = B[(long)k * N + n0 + c];
  }
  __syncthreads();

  const int wave = tid >> 5;
  const int lane = tid & 31;
  const int sub  = lane >> 4;   // 0: lanes 0-15, 1: lanes 16-31
  const int l16  = lane & 15;

  const long row0 = (long)blockIdx.x * 128 + wave * 16;
  long arow = row0 + l16;
  const long mr = (long)M - 1;
  if (arow > mr) arow = mr;                      // clamp, never predicate
  const float* __restrict__ Arow = A + arow * 256 + sub * 2;

  v8f acc0 = {};
  v8f acc1 = {};
  #pragma unroll 8
  for (int k = 0; k < 256; k += 4) {
    const v2f a  = *(const v2f*)(Arow + k);                       // {ks, ks+1}
    const int kp = (k >> 1) + sub;                                // B k-pair
    const v2f b0 = *(const v2f*)(Blds + kp * 64 + l16 * 2);       // cols n0+l16
    const v2f b1 = *(const v2f*)(Blds + kp * 64 + (l16 + 16) * 2);// cols n0+16+l16
    acc0 = __builtin_amdgcn_wmma_f32_16x16x4_f32(
        false, a, false, b0, (short)0, acc0, false, false);
    acc1 = __builtin_amdgcn_wmma_f32_16x16x4_f32(
        false, a, false, b1, (short)0, acc1, false, false);
  }

  const float bv0 = bias[n0 + l16];
  const float bv1 = bias[n0 + 16 + l16];
  #pragma unroll
  for (int i = 0; i < 8; ++i) {
    const long r = row0 + sub * 8 + i;
    if (r < (long)M) {
      C[r * N + n0 + l16]      = acc0[i] + bv0;
      C[r * N + n0 + 16 + l16] = acc1[i] + bv1;
    }
  }
}

// ---------------------------------------------------------------------------
// Softmax over SUMP=16 logits per (b, q, h).  One thread per group.
// attn layout: [BS*LQ, HEADS*SUMP] row-major, normalized in place.
// ---------------------------------------------------------------------------
__global__ __launch_bounds__(256)
void msda_softmax(float* __restrict__ attn, int total) {
  const int gid = blockIdx.x * blockDim.x + threadIdx.x;
  if (gid >= total) return;
  const int r = gid >> 3;
  const int h = gid & 7;
  float* p = attn + (long)r * (HEADS * SUMP) + h * SUMP;

  float m = p[0];
  #pragma unroll
  for (int i = 1; i < SUMP; ++i) m = fmaxf(m, p[i]);
  float e[SUMP];
  float s = 0.0f;
  #pragma unroll
  for (int i = 0; i < SUMP; ++i) { e[i] = expf(p[i] - m); s += e[i]; }
  const float inv = 1.0f / s;
  #pragma unroll
  for (int i = 0; i < SUMP; ++i) p[i] = e[i] * inv;
}

// ---------------------------------------------------------------------------
// Deformable sampling + attention-weighted sum.
// One wave per (b, h, q); the 32 lanes are the HEAD_DIM channels, so every
// corner gather is one coalesced 128B read of v[b][n][h*32 .. h*32+31].
// Output: head_out[(b*LQ+q)*256 + h*32 + d]  (== (bs, Lq, EMBED) layout).
// ---------------------------------------------------------------------------
__global__ __launch_bounds__(256)
void msda_sample(const float* __restrict__ v, const float* __restrict__ off,
                 const float* __restrict__ attn, const float* __restrict__ ref,
                 float* __restrict__ out) {
  const int wid  = blockIdx.x * 8 + (threadIdx.x >> 5);  // 0 .. BS*HEADS*LQ-1
  const int lane = threadIdx.x & 31;                     // channel d
  const int b    = wid / (HEADS * LQ);
  const int rem  = wid % (HEADS * LQ);
  const int h    = rem / LQ;
  const int q    = rem % LQ;

  const int   row = b * LQ + q;
  const float4 rp = *(const float4*)(ref + (long)row * 4);  // x, y, w, h
  const float* __restrict__ offp  = off  + (long)row * (HEADS * SUMP * 2) + h * (SUMP * 2);
  const float* __restrict__ attnp = attn + (long)row * (HEADS * SUMP)     + h * SUMP;
  const long vbase = ((long)b * LEN_V) * EMBED + h * HEAD_DIM + lane;

  const int lvl_start[LEVELS] = {0, 10000, 12500, 13125};
  const int lvl_hw[LEVELS]    = {100, 50, 25, 13};   // square levels

  float acc = 0.0f;
  #pragma unroll
  for (int p = 0; p < SUMP; ++p) {
    const int   l = p >> 2;
    const int   W = lvl_hw[l];
    const int   H = lvl_hw[l];
    const float ox = offp[2 * p];
    const float oy = offp[2 * p + 1];
    // loc = ref_xy + off * (1/NPTS) * ref_wh * 0.5 ;  x = loc*W - 0.5
    const float x = (rp.x + ox * rp.z * 0.125f) * (float)W - 0.5f;
    const float y = (rp.y + oy * rp.w * 0.125f) * (float)H - 0.5f;
    const float x0f = floorf(x), y0f = floorf(y);
    const float lx = x - x0f,    ly = y - y0f;
    const int   x0 = (int)x0f,   y0 = (int)y0f;
    const long  lb = vbase + (long)lvl_start[l] * EMBED;

    float s = 0.0f;
    #pragma unroll
    for (int c = 0; c < 4; ++c) {
      const int   dx = c & 1, dy = c >> 1;
      const float wx = dx ? lx : (1.0f - lx);
      const float wy = dy ? ly : (1.0f - ly);
      const int   xi = x0 + dx, yi = y0 + dy;
      const bool valid = (xi >= 0) && (xi < W) && (yi >= 0) && (yi < H);
      const int  xc = xi < 0 ? 0 : (xi > W - 1 ? W - 1 : xi);
      const int  yc = yi < 0 ? 0 : (yi > H - 1 ? H - 1 : yi);
      const float g = v[lb + (long)(yc * W + xc) * EMBED];
      s += g * (valid ? wx * wy : 0.0f);
    }
    acc += attnp[p] * s;
  }
  out[(long)row * EMBED + h * HEAD_DIM + lane] = acc;
}

// ---------------------------------------------------------------------------
extern "C" void kernel_launch(void* const* d_in, const int* in_sizes, int n_in,
                              void* d_out, int out_size, void* d_ws, size_t ws_size,
                              hipStream_t stream) {
  const float* query  = (const float*)d_in[0];
  const float* refpts = (const float*)d_in[1];
  const float* value  = (const float*)d_in[2];
  // d_in[3] = value_spatial_shapes (int32) -- compile-time constant here
  const float* W_v    = (const float*)d_in[4];
  const float* b_v    = (const float*)d_in[5];
  const float* W_off  = (const float*)d_in[6];
  const float* b_off  = (const float*)d_in[7];
  const float* W_attn = (const float*)d_in[8];
  const float* b_attn = (const float*)d_in[9];
  const float* W_out  = (const float*)d_in[10];
  const float* b_out  = (const float*)d_in[11];
  float* out = (float*)d_out;

  // Workspace carve-up (fp32 elements)
  float* ws      = (float*)d_ws;
  float* v_ws    = ws;                                   // [M_VAL, 256]   217.8 MB
  float* off_ws  = v_ws   + (size_t)M_VAL * EMBED;       // [M_Q, 256]
  float* attn_ws = off_ws + (size_t)M_Q * EMBED;         // [M_Q, 128]
  float* ho_ws   = attn_ws + (size_t)M_Q * (HEADS*SUMP); // [M_Q, 256]

  // 1) value projection: v = value @ W_v + b_v         -> [BS*LEN_V, 256]
  {
    dim3 grid((M_VAL + 127) / 128, EMBED / 32);
    gemm_wmma_f32<<<grid, 256, 0, stream>>>(value, W_v, b_v, v_ws, M_VAL, EMBED);
  }
  // 2) offsets: off = query @ W_off + b_off            -> [4800, 256]
  {
    dim3 grid((M_Q + 127) / 128, EMBED / 32);
    gemm_wmma_f32<<<grid, 256, 0, stream>>>(query, W_off, b_off, off_ws, M_Q, EMBED);
  }
  // 3) attention logits: query @ W_attn + b_attn       -> [4800, 128]
  {
    dim3 grid((M_Q + 127) / 128, (HEADS * SUMP) / 32);
    gemm_wmma_f32<<<grid, 256, 0, stream>>>(query, W_attn, b_attn, attn_ws, M_Q, HEADS * SUMP);
  }
  // 4) softmax over SUMP per (b,q,h)
  {
    const int total = M_Q * HEADS;                      // 38400
    msda_softmax<<<(total + 255) / 256, 256, 0, stream>>>(attn_ws, total);
  }
  // 5) bilinear sampling + weighted head sum           -> [4800, 256]
  {
    msda_sample<<<M_Q, 256, 0, stream>>>(v_ws, off_ws, attn_ws, refpts, ho_ws);
  }
  // 6) output projection: out = head_out @ W_out + b_out -> d_out [16,300,256]
  {
    dim3 grid((M_Q + 127) / 128, EMBED / 32);
    gemm_wmma_f32<<<grid, 256, 0, stream>>>(ho_ws, W_out, b_out, out, M_Q, EMBED);
  }
}